// DSSConvExcl_3736621547803
// MI455X (gfx1250) — compile-verified
//
#include <hip/hip_runtime.h>

#define NNODES 100000
#define RREL 4
#define DD 64
#define NR (NNODES * RREL)      // 400000 rows
#define ROWBLOCKS (NR / 16)     // 25000 16-row blocks
#define WST 68                  // padded LDS stride (conflict-free, 8B aligned)

typedef __attribute__((ext_vector_type(2))) float v2f;
typedef __attribute__((ext_vector_type(4))) float v4f;
typedef __attribute__((ext_vector_type(8))) float v8f;

// ---------------------------------------------------------------------------
// Kernel 1: H1[n,r,:] = vfts[n,r,:] ; H2[n,r,:] = (total[n]-vfts[n,r,:])/3
// Also zero the BN stats buffer (128 floats).
// ---------------------------------------------------------------------------
__global__ void init_kernel(const float* __restrict__ vfts,
                            float* __restrict__ H1, float* __restrict__ H2,
                            float* __restrict__ stats) {
  int idx = blockIdx.x * blockDim.x + threadIdx.x;
  if (idx < 128) stats[idx] = 0.0f;
  if (idx >= NNODES * 16) return;
  int n = idx >> 4;
  int c = idx & 15;               // 16 float4 chunks cover D=64
  const float* base = vfts + (size_t)n * RREL * DD + c * 4;
  v4f v0 = *(const v4f*)(base + 0 * DD);
  v4f v1 = *(const v4f*)(base + 1 * DD);
  v4f v2 = *(const v4f*)(base + 2 * DD);
  v4f v3 = *(const v4f*)(base + 3 * DD);
  v4f T = v0 + v1 + v2 + v3;
  const float k = 1.0f / 3.0f;
  float* h1 = H1 + (size_t)n * RREL * DD + c * 4;
  float* h2 = H2 + (size_t)n * RREL * DD + c * 4;
  *(v4f*)(h1 + 0 * DD) = v0;
  *(v4f*)(h1 + 1 * DD) = v1;
  *(v4f*)(h1 + 2 * DD) = v2;
  *(v4f*)(h1 + 3 * DD) = v3;
  *(v4f*)(h2 + 0 * DD) = (T - v0) * k;
  *(v4f*)(h2 + 1 * DD) = (T - v1) * k;
  *(v4f*)(h2 + 2 * DD) = (T - v2) * k;
  *(v4f*)(h2 + 3 * DD) = (T - v3) * k;
}

// ---------------------------------------------------------------------------
// Kernel 2: edge scatter. 16 lanes per edge, float4 per lane.
//   H1[dst,q]  += vfts[src,q]                       (q = rel of edge)
//   H2[dst,r]  += (total[src]-vfts[src,r])/3        (for r != q)
// ---------------------------------------------------------------------------
__global__ void edge_kernel(const float* __restrict__ vfts,
                            const long long* __restrict__ adjs,
                            const long long* __restrict__ rels,
                            float* __restrict__ H1, float* __restrict__ H2,
                            int E) {
  int gid = blockIdx.x * blockDim.x + threadIdx.x;
  int e = gid >> 4;
  int c = gid & 15;
  if (e >= E) return;
  int src = (int)adjs[e];
  int dst = (int)adjs[(size_t)E + e];
  int q   = (int)rels[e];
  const float* base = vfts + (size_t)src * RREL * DD + c * 4;
  v4f v[4];
  v[0] = *(const v4f*)(base + 0 * DD);
  v[1] = *(const v4f*)(base + 1 * DD);
  v[2] = *(const v4f*)(base + 2 * DD);
  v[3] = *(const v4f*)(base + 3 * DD);
  v4f T = v[0] + v[1] + v[2] + v[3];
  const float k = 1.0f / 3.0f;
#pragma unroll
  for (int r = 0; r < 4; ++r) {
    if (r == q) {
      float* hp = H1 + ((size_t)dst * RREL + r) * DD + c * 4;
      atomicAdd(hp + 0, v[r].x);
      atomicAdd(hp + 1, v[r].y);
      atomicAdd(hp + 2, v[r].z);
      atomicAdd(hp + 3, v[r].w);
    } else {
      v4f val = (T - v[r]) * k;
      float* hp = H2 + ((size_t)dst * RREL + r) * DD + c * 4;
      atomicAdd(hp + 0, val.x);
      atomicAdd(hp + 1, val.y);
      atomicAdd(hp + 2, val.z);
      atomicAdd(hp + 3, val.w);
    }
  }
}

// ---------------------------------------------------------------------------
// WMMA 2-layer MLP on one 16-row block: res = relu(A@Wa^T + ba)@Wb^T + bb
// A fragments (fp32 16x16x4): lane L holds m=L%16, k pair {4j+2hi, 4j+2hi+1}.
// B fragments: lane L holds n=L%16(+16t), same k pairing (W is row-major [n][k]).
// C/D: vgpr v, lane L -> element (m = v+8*hi, n = L%16 + 16t).
// ---------------------------------------------------------------------------
template <bool ACCUM>
__device__ inline void mlp16(const v2f* aIn, const float* lwa, const float* lba,
                             const float* lwb, const float* lbb, float* scr,
                             int m16, int hi, v8f* res) {
  // Stage A: relu(A @ Wa^T + ba) -> LDS scratch (16 x 64, stride WST)
#pragma unroll
  for (int t = 0; t < 4; ++t) {
    int n = t * 16 + m16;
    float bias = lba[n];
    v8f acc;
#pragma unroll
    for (int v = 0; v < 8; ++v) acc[v] = bias;
    const float* wn = lwa + n * WST;
#pragma unroll
    for (int j = 0; j < 16; ++j) {
      v2f b = *(const v2f*)(wn + 4 * j + 2 * hi);
      acc = __builtin_amdgcn_wmma_f32_16x16x4_f32(false, aIn[j], false, b,
                                                  (short)0, acc, false, false);
    }
#pragma unroll
    for (int v = 0; v < 8; ++v) {
      float y = acc[v] > 0.0f ? acc[v] : 0.0f;
      scr[(v + 8 * hi) * WST + n] = y;   // D-layout -> row-major in LDS
    }
  }
  // Stage B A-fragments from scratch (row-major -> A layout)
  v2f ab[16];
#pragma unroll
  for (int j = 0; j < 16; ++j)
    ab[j] = *(const v2f*)(scr + m16 * WST + 4 * j + 2 * hi);
#pragma unroll
  for (int t = 0; t < 4; ++t) {
    int n = t * 16 + m16;
    float bias = lbb[n];
    v8f acc;
#pragma unroll
    for (int v = 0; v < 8; ++v) acc[v] = bias;
    const float* wn = lwb + n * WST;
#pragma unroll
    for (int j = 0; j < 16; ++j) {
      v2f b = *(const v2f*)(wn + 4 * j + 2 * hi);
      acc = __builtin_amdgcn_wmma_f32_16x16x4_f32(false, ab[j], false, b,
                                                  (short)0, acc, false, false);
    }
    if (ACCUM) res[t] = res[t] + acc; else res[t] = acc;
  }
}

// ---------------------------------------------------------------------------
// Kernel 3: out = MLP1(H1) + MLP2(H2), plus per-feature sum/sumsq for BN.
// 8 waves per block, one 16-row block per wave per grid-stride step.
// ---------------------------------------------------------------------------
__global__ __launch_bounds__(256) void mlp_kernel(
    const float* __restrict__ H1, const float* __restrict__ H2,
    const float* __restrict__ w1a, const float* __restrict__ b1a,
    const float* __restrict__ w1b, const float* __restrict__ b1b,
    const float* __restrict__ w2a, const float* __restrict__ b2a,
    const float* __restrict__ w2b, const float* __restrict__ b2b,
    float* __restrict__ out, float* __restrict__ stats) {
  __shared__ float sm[4 * 64 * WST + 256 + 8 * 16 * WST];  // ~105 KB (CDNA5 WGP LDS)
  float* lw1a = sm;
  float* lw1b = sm + 1 * 64 * WST;
  float* lw2a = sm + 2 * 64 * WST;
  float* lw2b = sm + 3 * 64 * WST;
  float* lb   = sm + 4 * 64 * WST;            // b1a | b1b | b2a | b2b
  float* lscr = sm + 4 * 64 * WST + 256;

  int tid = threadIdx.x;
  for (int i = tid; i < 64 * 64; i += 256) {
    int n = i >> 6, k = i & 63;
    lw1a[n * WST + k] = w1a[i];
    lw1b[n * WST + k] = w1b[i];
    lw2a[n * WST + k] = w2a[i];
    lw2b[n * WST + k] = w2b[i];
  }
  if (tid < 64) {
    lb[tid]       = b1a[tid];
    lb[64 + tid]  = b1b[tid];
    lb[128 + tid] = b2a[tid];
    lb[192 + tid] = b2b[tid];
  }
  __syncthreads();

  int lane = tid & 31;
  int wave = tid >> 5;
  int m16 = lane & 15;
  int hi = lane >> 4;
  float* scr = lscr + wave * 16 * WST;

  int gwave = blockIdx.x * 8 + wave;
  int nWaves = gridDim.x * 8;

  float sAcc[4] = {0.f, 0.f, 0.f, 0.f};
  float ssAcc[4] = {0.f, 0.f, 0.f, 0.f};

  for (int rb = gwave; rb < ROWBLOCKS; rb += nWaves) {
    int row0 = rb * 16;
    const float* r1 = H1 + (size_t)(row0 + m16) * DD;
    const float* r2 = H2 + (size_t)(row0 + m16) * DD;
    v2f a1[16], a2[16];
#pragma unroll
    for (int j = 0; j < 16; ++j) {
      a1[j] = *(const v2f*)(r1 + 4 * j + 2 * hi);
      a2[j] = *(const v2f*)(r2 + 4 * j + 2 * hi);
    }

    v8f res[4];
    mlp16<false>(a1, lw1a, lb, lw1b, lb + 64, scr, m16, hi, res);
    mlp16<true>(a2, lw2a, lb + 128, lw2b, lb + 192, scr, m16, hi, res);

#pragma unroll
    for (int t = 0; t < 4; ++t) {
      int n = t * 16 + m16;
#pragma unroll
      for (int v = 0; v < 8; ++v) {
        float val = res[t][v];
        out[(size_t)(row0 + v + 8 * hi) * DD + n] = val;
        sAcc[t] += val;
        ssAcc[t] += val * val;
      }
    }
  }

  // lanes L and L+16 share the same feature column: pairwise reduce, then atomics
#pragma unroll
  for (int t = 0; t < 4; ++t) {
    float s = sAcc[t] + __shfl_xor(sAcc[t], 16, 32);
    float ss = ssAcc[t] + __shfl_xor(ssAcc[t], 16, 32);
    if (hi == 0) {
      atomicAdd(&stats[t * 16 + m16], s);
      atomicAdd(&stats[64 + t * 16 + m16], ss);
    }
  }
}

// ---------------------------------------------------------------------------
// Kernel 4: batch-norm normalize in place on d_out.
// ---------------------------------------------------------------------------
__global__ void bn_kernel(float* __restrict__ out, const float* __restrict__ stats,
                          const float* __restrict__ gamma,
                          const float* __restrict__ beta) {
  int idx = blockIdx.x * blockDim.x + threadIdx.x;
  if (idx >= NR * DD / 4) return;
  int c4 = (idx & 15) * 4;
  v4f x = *(v4f*)(out + (size_t)idx * 4);
  const float invM = 1.0f / (float)NR;
#pragma unroll
  for (int i = 0; i < 4; ++i) {
    int c = c4 + i;
    float mean = stats[c] * invM;
    float var = stats[64 + c] * invM - mean * mean;
    float inv = rsqrtf(var + 1e-5f);
    x[i] = (x[i] - mean) * inv * gamma[c] + beta[c];
  }
  *(v4f*)(out + (size_t)idx * 4) = x;
}

// ---------------------------------------------------------------------------
extern "C" void kernel_launch(void* const* d_in, const int* in_sizes, int n_in,
                              void* d_out, int out_size, void* d_ws, size_t ws_size,
                              hipStream_t stream) {
  const float* vfts = (const float*)d_in[0];
  const long long* adjs = (const long long*)d_in[1];
  const long long* rels = (const long long*)d_in[2];
  const float* w1a = (const float*)d_in[3];
  const float* b1a = (const float*)d_in[4];
  const float* w1b = (const float*)d_in[5];
  const float* b1b = (const float*)d_in[6];
  const float* w2a = (const float*)d_in[7];
  const float* b2a = (const float*)d_in[8];
  const float* w2b = (const float*)d_in[9];
  const float* b2b = (const float*)d_in[10];
  const float* gamma = (const float*)d_in[11];
  const float* beta = (const float*)d_in[12];
  float* out = (float*)d_out;
  int E = in_sizes[2];

  float* H1 = (float*)d_ws;
  float* H2 = H1 + (size_t)NR * DD;
  float* stats = H2 + (size_t)NR * DD;

  init_kernel<<<(NNODES * 16 + 255) / 256, 256, 0, stream>>>(vfts, H1, H2, stats);

  unsigned edgeBlocks = (unsigned)(((size_t)E * 16 + 255) / 256);
  edge_kernel<<<edgeBlocks, 256, 0, stream>>>(vfts, adjs, rels, H1, H2, E);

  mlp_kernel<<<625, 256, 0, stream>>>(H1, H2, w1a, b1a, w1b, b1b,
                                      w2a, b2a, w2b, b2b, out, stats);

  bn_kernel<<<(NR * DD / 4 + 255) / 256, 256, 0, stream>>>(out, stats, gamma, beta);
}